// AbstractModel_70394513981830
// MI455X (gfx1250) — compile-verified
//
#include <hip/hip_runtime.h>

#define THETA_F   1.0f
#define BLOCK     256
#define EPT       4                    // edges per thread per chunk (B128 loads)
#define CHUNK     (BLOCK * EPT)        // 1024 edges per chunk
#define N_STEPS   200
#define EQ_STEPS  100

// ---------------- CDNA5 async global->LDS gather --------------------------
#if defined(__has_builtin)
#  if __has_builtin(__builtin_amdgcn_global_load_async_to_lds_b32)
#    define HAVE_ASYNC_LDS 1
#  endif
#endif
#ifndef HAVE_ASYNC_LDS
#  define HAVE_ASYNC_LDS 0
#endif

typedef __attribute__((address_space(1))) int global_i32;   // global (AS1)
typedef __attribute__((address_space(3))) int lds_i32;      // LDS (AS3)

__device__ __forceinline__ void async_gather_b32(const float* g, float* l) {
#if HAVE_ASYNC_LDS
  // GLOBAL_LOAD_ASYNC_TO_LDS_B32: per-lane VGPR global address -> per-lane LDS
  // slot, tracked by ASYNCcnt (separate from LOADcnt).
  __builtin_amdgcn_global_load_async_to_lds_b32(
      (global_i32*)g,
      (lds_i32*)l,
      /*imm offset*/0, /*cpol*/0);
#else
  *l = *g;   // synchronous fallback: gather through a register into LDS
#endif
}

__device__ __forceinline__ void wait_async_le(int pending_keep) {
  if (pending_keep == 0)      __builtin_amdgcn_s_wait_asynccnt(0);
  else                        __builtin_amdgcn_s_wait_asynccnt(EPT);
  __asm__ volatile("" ::: "memory");   // keep LDS read after the wait
}

__device__ __forceinline__ void atomic_add_f32(float* p, float v) {
  // Lowers to global_atomic_add_f32 (no return) executed at L2 on CDNA.
  unsafeAtomicAdd(p, v);
}

// ---------------- edge scatter: acc[dst] += W * s[src] ---------------------
__global__ __launch_bounds__(BLOCK) void edge_scatter_kernel(
    const int* __restrict__ src, const int* __restrict__ dst,
    const float* __restrict__ W, const float* __restrict__ s,
    float* __restrict__ acc, int n_edges, int nchunks)
{
  __shared__ __align__(16) float sbuf[2][CHUNK];
  const int tid = threadIdx.x;

  int c = blockIdx.x;
  // Prologue: issue gathers for this block's first chunk (block-uniform branch,
  // EXEC is all-ones for the async ops).
  if (c < nchunks) {
    const int base = c * CHUNK + tid * EPT;
    int4 s4 = *(const int4*)(src + base);
    async_gather_b32(s + s4.x, &sbuf[0][tid * EPT + 0]);
    async_gather_b32(s + s4.y, &sbuf[0][tid * EPT + 1]);
    async_gather_b32(s + s4.z, &sbuf[0][tid * EPT + 2]);
    async_gather_b32(s + s4.w, &sbuf[0][tid * EPT + 3]);
  }

  int buf = 0;
  for (; c < nchunks; c += gridDim.x) {
    const int cn = c + gridDim.x;
    if (cn < nchunks) {
      // Issue next chunk's gathers into the other buffer, prefetch its
      // streaming edge data, then wait until only those EPT ops are in flight.
      const int nbase = cn * CHUNK + tid * EPT;
      int4 s4 = *(const int4*)(src + nbase);
      __builtin_prefetch(W + nbase, 0, 0);     // global_prefetch_b8
      __builtin_prefetch(dst + nbase, 0, 0);
      async_gather_b32(s + s4.x, &sbuf[buf ^ 1][tid * EPT + 0]);
      async_gather_b32(s + s4.y, &sbuf[buf ^ 1][tid * EPT + 1]);
      async_gather_b32(s + s4.z, &sbuf[buf ^ 1][tid * EPT + 2]);
      async_gather_b32(s + s4.w, &sbuf[buf ^ 1][tid * EPT + 3]);
      wait_async_le(1);
    } else {
      wait_async_le(0);
    }

    // Consume current chunk: per-lane ds_load_b128 of its own slot.
    float4 sv = *(const float4*)&sbuf[buf][tid * EPT];
    const int base = c * CHUNK + tid * EPT;
    float4 w4 = *(const float4*)(W + base);
    int4  d4 = *(const int4*)(dst + base);
    atomic_add_f32(acc + d4.x, w4.x * sv.x);
    atomic_add_f32(acc + d4.y, w4.y * sv.y);
    atomic_add_f32(acc + d4.z, w4.z * sv.z);
    atomic_add_f32(acc + d4.w, w4.w * sv.w);
    buf ^= 1;
  }

  // Generic tail (empty for 6.4M edges: 6400000 % 1024 == 0).
  for (int e = nchunks * CHUNK + blockIdx.x * BLOCK + tid; e < n_edges;
       e += gridDim.x * BLOCK) {
    atomic_add_f32(acc + dst[e], W[e] * s[src[e]]);
  }
}

// ---------------- node update: s = sigmoid(acc - theta); acc = 0 ----------
__global__ __launch_bounds__(256) void node_update_kernel(
    float* __restrict__ acc, float* __restrict__ s,
    float* __restrict__ out_row, int n)
{
  const int i = blockIdx.x * 256 + threadIdx.x;
  if (i < n) {
    const float inp = acc[i];
    acc[i] = 0.0f;                       // ready for next step's atomics
    const float e = __expf(THETA_F - inp);     // v_exp_f32
    const float r = __frcp_rn(1.0f + e);       // sigmoid(inp - theta)
    s[i] = r;
    if (out_row) out_row[i] = r;
  }
}

// ---------------- init: s = x, acc = 0 ------------------------------------
__global__ __launch_bounds__(256) void init_kernel(
    const float* __restrict__ x, float* __restrict__ s,
    float* __restrict__ acc, int n)
{
  const int i = blockIdx.x * 256 + threadIdx.x;
  if (i < n) {
    s[i] = x[i];
    acc[i] = 0.0f;
  }
}

extern "C" void kernel_launch(void* const* d_in, const int* in_sizes, int n_in,
                              void* d_out, int out_size, void* d_ws, size_t ws_size,
                              hipStream_t stream) {
  (void)n_in; (void)out_size; (void)ws_size;
  const float* x  = (const float*)d_in[0];          // (N_NODES, 1)
  const float* W  = (const float*)d_in[1];          // (N_EDGES,)
  const int*   ei = (const int*)d_in[2];            // (2, N_EDGES): row0=src row1=dst
  const int n_nodes = in_sizes[0];
  const int n_edges = in_sizes[1];
  const int* src = ei;
  const int* dst = ei + n_edges;

  float* s   = (float*)d_ws;                        // n_nodes floats
  float* acc = s + n_nodes;                         // n_nodes floats
  float* out = (float*)d_out;                       // (N_STEPS, n_nodes)

  const int nchunks = n_edges / CHUNK;              // 6250
  int egrid = nchunks < 2048 ? (nchunks > 0 ? nchunks : 1) : 2048; // persistent
  const int ngrid = (n_nodes + 255) / 256;

  init_kernel<<<ngrid, 256, 0, stream>>>(x, s, acc, n_nodes);

  const int total = EQ_STEPS + N_STEPS;             // 300 sequential steps
  for (int it = 0; it < total; ++it) {
    edge_scatter_kernel<<<egrid, BLOCK, 0, stream>>>(src, dst, W, s, acc,
                                                     n_edges, nchunks);
    float* row = (it >= EQ_STEPS) ? out + (size_t)(it - EQ_STEPS) * n_nodes
                                  : nullptr;
    node_update_kernel<<<ngrid, 256, 0, stream>>>(acc, s, row, n_nodes);
  }
}